// NegativeActivationLayer_12936441496125
// MI455X (gfx1250) — compile-verified
//
#include <hip/hip_runtime.h>

typedef __attribute__((ext_vector_type(2))) float v2f;
typedef __attribute__((ext_vector_type(8))) float v8f;

#define LOG2E 1.44269504088896340736f
#define LN2   0.69314718055994530942f

// workspace layout (float offsets)
#define CN_OFF 0          // [16 o_tiles][256 i][16 olo][4]  (T1, B2, B3, B4)   = 262144 floats
#define PB_OFF 262144     // [16 o_tiles][256 i][32 lane][2] (WMMA B operand)   = 262144 floats
#define C0_OFF 524288     // [256 o][256 i] constant contributions               = 65536 floats
#define K0_OFF 589824     // [256 o] reduced constants                           = 256 floats

// ---------------------------------------------------------------------------
// Kernel 1: per-(o,i) coefficient precompute (splines, tanh, softplus)
// ---------------------------------------------------------------------------
__global__ void precompute_kernel(const float* __restrict__ raw_gamma,
                                  const float* __restrict__ w,
                                  const float* __restrict__ breaks,
                                  const float* __restrict__ coefs,
                                  const float* __restrict__ mu,
                                  const float* __restrict__ sigma,
                                  float* __restrict__ ws) {
  const int o = blockIdx.x;   // 0..255
  const int i = threadIdx.x;  // 0..255

  float wv = w[o * 256 + i];
  wv = fminf(fmaxf(wv, -5.5f), 37.9f);
  const float wn = (wv - mu[0]) / sigma[0];

  float b[8];
#pragma unroll
  for (int s = 0; s < 8; ++s) {
    const float* br = breaks + s * 33;
    const float* cf = coefs + s * 32 * 4;
    const float lo = br[0];
    const float hi = br[32] - 1e-6f;
    float wc = fminf(fmaxf(wn, lo), hi);
    // searchsorted(br, wc, 'left'): first j with br[j] >= wc
    int k = 33;
    for (int j = 0; j < 33; ++j) {
      if (br[j] >= wc) { k = j; break; }
    }
    int idx = k - 1;
    idx = idx < 0 ? 0 : (idx > 31 ? 31 : idx);
    const float t = wc - br[idx];
    const float* a = cf + idx * 4;
    b[s] = ((a[0] * t + a[1]) * t + a[2]) * t + a[3];
  }

  const float th = tanhf(10.0f * wn);
  const float rg = raw_gamma[o * 256 + i];
  const float sp = (rg > 0.0f) ? (rg + log1pf(expf(-rg))) : log1pf(expf(rg));
  const float g  = sp * (0.01f / 256.0f);   // gamma * fixed_downscale
  const float tg = th * g;

  const int o_tile = o >> 4;
  const int olo    = o & 15;

  // nonlinear coefs: folded ln2/log2e constants are applied in main kernel
  float* CN = ws + CN_OFF + (size_t)((o_tile * 256 + i) * 16 + olo) * 4;
  CN[0] = tg * b[0];   // T1
  CN[1] = b[1];        // B2
  CN[2] = b[2];        // B3
  CN[3] = b[3];        // B4

  // WMMA B operand layout: lane l = h*16+olo holds {P(2h+1), P(2h+2)}
  float* PB = ws + PB_OFF + (size_t)((o_tile * 256 + i) * 32) * 2;
  PB[olo * 2 + 0]        = tg * b[4];  // P1 (x^1 weight), K=0 row
  PB[olo * 2 + 1]        = tg * b[5];  // P2 (x^2 weight), K=1 row
  PB[(16 + olo) * 2 + 0] = tg * b[6];  // P3 (x^3 weight), K=2 row
  PB[(16 + olo) * 2 + 1] = tg * b[7];  // P4 (x^4 weight), K=3 row

  // constant contribution (independent of b)
  ws[C0_OFF + o * 256 + i] = g * 19.0f * (1.0f - th);
}

// ---------------------------------------------------------------------------
// Kernel 2: K0[o] = sum_i C0[o,i]  (deterministic reduction)
// ---------------------------------------------------------------------------
__global__ void k0_kernel(float* __restrict__ ws) {
  const int o = threadIdx.x;
  const float* C0 = ws + C0_OFF + o * 256;
  float s = 0.0f;
  for (int i = 0; i < 256; ++i) s += C0[i];
  ws[K0_OFF + o] = s;
}

// ---------------------------------------------------------------------------
// Kernel 3: main fused kernel. Block = 128 threads = 4 waves, one 16x16 tile.
// Waves K-split IN=256 into 4 ranges of 64. Per i: one WMMA f32 16x16x4 for
// the polynomial GEMM + 8 transcendental chains per lane for the nonlinear sum.
// ---------------------------------------------------------------------------
__global__ void __launch_bounds__(128) main_kernel(const float* __restrict__ x,
                                                   const float* __restrict__ ws,
                                                   float* __restrict__ out) {
  const int o_tile = blockIdx.x & 15;   // 16 o-tiles
  const int b_tile = blockIdx.x >> 4;   // 64 b-tiles
  const int lane   = threadIdx.x & 31;
  const int wvid   = threadIdx.x >> 5;  // wave 0..3
  const int half   = lane >> 4;
  const int olo    = lane & 15;

  const int arow    = b_tile * 16 + olo;            // A-operand row (M = lane%16)
  const int rowbase = b_tile * 16 + half * 8;       // nonlinear rows for this lane

  const float* CN = ws + CN_OFF;
  const float* PB = ws + PB_OFF;

  v8f c = {0.f, 0.f, 0.f, 0.f, 0.f, 0.f, 0.f, 0.f};
  float nl[8];
#pragma unroll
  for (int r = 0; r < 8; ++r) nl[r] = 0.0f;

  const int i0 = wvid * 64;
  for (int ii = 0; ii < 64; ++ii) {
    const int i = i0 + ii;

    // --- loads (all L0/L2-resident; total footprint ~3 MB) ---
    const float  xa = fmaxf(x[arow * 256 + i], 0.0f);
    const float4 cn = *(const float4*)(CN + (size_t)((o_tile * 256 + i) * 16 + olo) * 4);
    const float2 pb = *(const float2*)(PB + (size_t)((o_tile * 256 + i) * 32 + lane) * 2);

    // --- polynomial GEMM slice via WMMA f32 16x16x4 ---
    // A 16x4 layout: lanes 0-15 hold {K0,K1} = {x, x^2}; lanes 16-31 {K2,K3} = {x^3, x^4}
    const float x2 = xa * xa;
    v2f A, B;
    A.x = half ? x2 * xa : xa;
    A.y = half ? x2 * x2 : x2;
    B.x = pb.x;
    B.y = pb.y;
    c = __builtin_amdgcn_wmma_f32_16x16x4_f32(false, A, false, B, (short)0, c,
                                              false, false);

    // --- nonlinear term: T1*ln(1 + B2*ln(1 + (e^{B3*x}-1)^{B4})) for 8 rows ---
    const float b3l = cn.z * LOG2E;  // fold e^x -> 2^x
    const float b2l = cn.y * LN2;    // fold ln -> log2
    const float t1l = cn.x * LN2;    // fold ln -> log2
    const float b4v = cn.w;
    const float* xrow = x + (size_t)rowbase * 256 + i;
#pragma unroll
    for (int r = 0; r < 8; ++r) {
      const float xv    = fmaxf(xrow[r * 256], 0.0f);
      const float e     = __builtin_amdgcn_exp2f(b3l * xv);        // v_exp_f32
      const float base  = e - 1.0f;                                // >= 0 (b3 > 0)
      const float p     = __builtin_amdgcn_exp2f(b4v * __builtin_amdgcn_logf(base));
      const float inner = fmaf(b2l, __builtin_amdgcn_logf(1.0f + p), 1.0f);
      nl[r] = fmaf(t1l, __builtin_amdgcn_logf(inner), nl[r]);
    }
  }

  // --- cross-wave reduction (K-split) through LDS ---
  __shared__ float red[4 * 8 * 32];
#pragma unroll
  for (int r = 0; r < 8; ++r) red[(wvid * 8 + r) * 32 + lane] = c[r] + nl[r];
  __syncthreads();

  for (int t = threadIdx.x; t < 256; t += 128) {
    const int bl = t >> 4;              // local b row 0..15
    const int ol = t & 15;              // local o col
    const int lane2 = (bl >> 3) * 16 + ol;
    const int r2    = bl & 7;
    float s = ws[K0_OFF + o_tile * 16 + ol];
#pragma unroll
    for (int q = 0; q < 4; ++q) s += red[(q * 8 + r2) * 32 + lane2];
    out[(size_t)(b_tile * 16 + bl) * 256 + (o_tile * 16 + ol)] = s;
  }
}

// ---------------------------------------------------------------------------
extern "C" void kernel_launch(void* const* d_in, const int* in_sizes, int n_in,
                              void* d_out, int out_size, void* d_ws, size_t ws_size,
                              hipStream_t stream) {
  const float* x         = (const float*)d_in[0];
  const float* raw_gamma = (const float*)d_in[1];
  const float* w         = (const float*)d_in[2];
  const float* breaks    = (const float*)d_in[3];
  const float* coefs     = (const float*)d_in[4];
  const float* mu        = (const float*)d_in[5];
  const float* sigma     = (const float*)d_in[6];
  float* ws  = (float*)d_ws;
  float* out = (float*)d_out;

  precompute_kernel<<<256, 256, 0, stream>>>(raw_gamma, w, breaks, coefs, mu, sigma, ws);
  k0_kernel<<<1, 256, 0, stream>>>(ws);
  main_kernel<<<1024, 128, 0, stream>>>(x, ws, out);
}